// OutputLayer_71966472012426
// MI455X (gfx1250) — compile-verified
//
#include <hip/hip_runtime.h>
#include <math.h>

// ---------------------------------------------------------------------------
// MI455X / gfx1250 implementation.
// One wave32 per pair. FP32 WMMA (v_wmma_f32_16x16x4_f32) for the two GEMMs:
//   step A:  D_l = gw_l(16x16) @ Cin_l(16x(2l+1))        -> 20 WMMAs/pair
//   step B:  M   = CGall(58x25, constant) @ CoutT(25x16) -> 28 WMMAs/pair
// CG tables (numpy RandomState MT19937 + legacy gauss) generated on device
// into d_ws by a setup kernel.
// ---------------------------------------------------------------------------

typedef __attribute__((ext_vector_type(2))) float v2f;
typedef __attribute__((ext_vector_type(8))) float v8f;

#define WAVES_PER_BLOCK 8
#define BLOCK_THREADS   (WAVES_PER_BLOCK * 32)

#define CG_ROWS 64   // 58 valid rows, padded
#define CG_COLS 28   // 25 valid cols (l3,a), padded to multiple of 4
#define CG_SIZE (CG_ROWS * CG_COLS)

// per-wave LDS scratch layout (floats)
#define RBF_OFF     0            // 32
#define G_OFF       32           // 5*16 = 80
#define COUT_OFF    112          // 28*16 = 448  (CoutT, K-padded)
#define M_OFF       560          // 64*16 = 1024 (M rows, 58 valid)
#define WAVE_FLOATS 1584

#define WMMA_F32X4(A, B, C) \
  __builtin_amdgcn_wmma_f32_16x16x4_f32(false, (A), false, (B), (short)0, (C), false, false)

// ------------------- MT19937 + numpy legacy gauss --------------------------
__device__ static unsigned mt_next(unsigned* mt, int& mti) {
  if (mti >= 624) {
    for (int k = 0; k < 624; ++k) {
      unsigned y = (mt[k] & 0x80000000u) | (mt[(k + 1) % 624] & 0x7fffffffu);
      unsigned v = mt[(k + 397) % 624] ^ (y >> 1);
      if (y & 1u) v ^= 0x9908b0dfu;
      mt[k] = v;
    }
    mti = 0;
  }
  unsigned y = mt[mti++];
  y ^= (y >> 11);
  y ^= (y << 7)  & 0x9d2c5680u;
  y ^= (y << 15) & 0xefc60000u;
  y ^= (y >> 18);
  return y;
}

__device__ static double mt_double(unsigned* mt, int& mti) {
  unsigned a = mt_next(mt, mti) >> 5;
  unsigned b = mt_next(mt, mti) >> 6;
  return ((double)a * 67108864.0 + (double)b) / 9007199254740992.0;
}

__device__ static double mt_gauss(unsigned* mt, int& mti, double& cache, int& has) {
  if (has) { has = 0; return cache; }
  double x1, x2, r2;
  do {
    x1 = 2.0 * mt_double(mt, mti) - 1.0;
    x2 = 2.0 * mt_double(mt, mti) - 1.0;
    r2 = x1 * x1 + x2 * x2;
  } while (r2 >= 1.0 || r2 == 0.0);
  double f = sqrt(-2.0 * log(r2) / r2);
  cache = f * x1; has = 1;
  return f * x2;
}

// Builds the padded 64x28 CGall matrix: row r = block(l1,l2) row (m,n),
// col = l3*l3 + a (concatenated (l3,a) index), zero elsewhere.
__global__ void cg_setup_kernel(float* __restrict__ cg) {
  __shared__ unsigned mt[624];
  for (int idx = threadIdx.x; idx < CG_SIZE; idx += blockDim.x) cg[idx] = 0.0f;
  __syncthreads();
  if (threadIdx.x != 0) return;
  const int base3[9] = {0, 0, 0, 1, 4, 0, 13, 18, 33}; // [l1*3+l2], l1>=l2
  for (int l1 = 0; l1 < 3; ++l1) {
    for (int l2 = 0; l2 <= l1; ++l2) {
      for (int l3 = l1 - l2; l3 <= l1 + l2; ++l3) {
        unsigned seed = (unsigned)(1000 + l1 * 100 + l2 * 10 + l3);
        mt[0] = seed;
        for (int k = 1; k < 624; ++k)
          mt[k] = 1812433253u * (mt[k - 1] ^ (mt[k - 1] >> 30)) + (unsigned)k;
        int mti = 624; double cache = 0.0; int has = 0;
        for (int mm = 0; mm < 2 * l1 + 1; ++mm)
          for (int nn = 0; nn < 2 * l2 + 1; ++nn)
            for (int aa = 0; aa < 2 * l3 + 1; ++aa) {
              float v = (float)mt_gauss(mt, mti, cache, has);
              int r = base3[l1 * 3 + l2] + mm * (2 * l2 + 1) + nn;
              cg[r * CG_COLS + l3 * l3 + aa] = v;
            }
      }
    }
  }
}

__device__ static inline void decode14(int x, int& l, int& q) {
  if (x < 3)      { l = 0; q = x; }
  else if (x < 9) { l = 1; q = x - 3; }
  else            { l = 2; q = x - 9; }
}

__launch_bounds__(BLOCK_THREADS)
__global__ void output_layer_kernel(
    const float* __restrict__ c0, const float* __restrict__ c1,
    const float* __restrict__ c2, const float* __restrict__ c3,
    const float* __restrict__ c4,
    const float* __restrict__ R,
    const float* __restrict__ w_rad,      // (5,32,16)
    const float* __restrict__ weight,     // (100,5,16,16)
    const float* __restrict__ s_col,      // (3,16)
    const float* __restrict__ p_col,      // (2,16)
    const float* __restrict__ d_col,      // (1,16)
    const int*   __restrict__ Z,
    const int*   __restrict__ pairs,      // (P,2)
    const int*   __restrict__ aidx,       // (P,)
    const float* __restrict__ cg,         // 64x28 from d_ws
    float*       __restrict__ out,        // (P,14,14)
    int P)
{
  __shared__ float s_cg[CG_SIZE];
  __shared__ float s_coll[96];            // s(48) | p(32) | d(16), each [a*16+f]
  __shared__ float s_wave[WAVES_PER_BLOCK][WAVE_FLOATS];

  for (int idx = threadIdx.x; idx < CG_SIZE; idx += BLOCK_THREADS) s_cg[idx] = cg[idx];
  for (int idx = threadIdx.x; idx < 96; idx += BLOCK_THREADS) {
    float v;
    if (idx < 48)      v = s_col[idx];
    else if (idx < 80) v = p_col[idx - 48];
    else               v = d_col[idx - 80];
    s_coll[idx] = v;
  }
  __syncthreads();

  const int lane = threadIdx.x & 31;
  const int wave = threadIdx.x >> 5;
  const int pair = blockIdx.x * WAVES_PER_BLOCK + wave;
  if (pair >= P) return;

  float* sw = s_wave[wave];

  const int i  = pairs[2 * pair + 0];
  const int j  = pairs[2 * pair + 1];
  const int ap = aidx[pair];
  const int t1 = Z[pairs[2 * ap + 0]];
  const int t2 = Z[pairs[2 * ap + 1]];

  const size_t wbase12 = (size_t)(t1 * 10 + t2) * 5 * 256;
  const size_t wbase21 = (size_t)(t2 * 10 + t1) * 5 * 256;
  __builtin_prefetch(weight + wbase12, 0, 0);   // global_prefetch_b8
  __builtin_prefetch(weight + wbase21, 0, 0);

  // ---- radial features: lane k handles basis k (K = 32 = wave width) ----
  const float PIF = 3.14159265358979323846f;
  const float dx = R[3 * i + 0] - R[3 * j + 0];
  const float dy = R[3 * i + 1] - R[3 * j + 1];
  const float dz = R[3 * i + 2] - R[3 * j + 2];
  const float d  = sqrtf(dx * dx + dy * dy + dz * dz);
  float env = 0.5f * (cosf(PIF * fminf(d, 5.0f) / 5.0f) + 1.0f);
  env = (d < 5.0f) ? env : 0.0f;
  sw[RBF_OFF + lane] = sinf((float)(lane + 1) * PIF * d / 5.0f) * env;

  // zero K-pad rows (25..27) of CoutT
  for (int idx = lane; idx < 48; idx += 32) sw[COUT_OFF + 25 * 16 + idx] = 0.0f;

  // g[l][c] = sum_k rbf[k] * w_rad[l][k][c]   (lanes 0..15 own c)
  {
    const int c = lane & 15;
#pragma unroll
    for (int l = 0; l < 5; ++l) {
      float s = 0.0f;
#pragma unroll
      for (int k = 0; k < 32; ++k) s += sw[RBF_OFF + k] * w_rad[(l * 32 + k) * 16 + c];
      if (lane < 16) sw[G_OFF + l * 16 + c] = s;
    }
  }

  const int m = lane & 15;   // WMMA: A row / B,C,D column index
  const int h = lane >> 4;   // half-wave -> K slots {0,1} vs {2,3}

  // ---- step A: per-l GEMM  D_l = gw_l(16x16) @ Cin_l(16x(2l+1)) ----
  const float* cls[5] = {c0, c1, c2, c3, c4};
#pragma unroll
  for (int l = 0; l < 5; ++l) {
    const int dim = 2 * l + 1;
    const float* cl = cls[l];
    v8f acc = {0.f, 0.f, 0.f, 0.f, 0.f, 0.f, 0.f, 0.f};
#pragma unroll
    for (int kk = 0; kk < 4; ++kk) {
      const int cch = kk * 4 + 2 * h;      // K channel for this lane half
      const float* w12 = weight + wbase12 + (size_t)l * 256 + m * 16 + cch;
      const float* w21 = weight + wbase21 + (size_t)l * 256 + m * 16 + cch;
      v2f a; a.x = w12[0] + w21[0]; a.y = w12[1] + w21[1];
      v2f b;
      if (m < dim) {
        const float g0 = sw[G_OFF + l * 16 + cch];
        const float g1 = sw[G_OFF + l * 16 + cch + 1];
        const size_t oi = ((size_t)i * 16 + cch) * dim + m;
        const size_t oj = ((size_t)j * 16 + cch) * dim + m;
        b.x = g0 * (cl[oi] + cl[oj]);
        b.y = g1 * (cl[oi + dim] + cl[oj + dim]);
      } else { b.x = 0.0f; b.y = 0.0f; }
      acc = WMMA_F32X4(a, b, acc);
    }
    // scatter D_l into CoutT LDS: CoutT[(l*l + m)][bchan] = D_l[bchan][m]
    if (m < dim) {
#pragma unroll
      for (int v = 0; v < 8; ++v) {
        const int bch = v + 8 * h;
        sw[COUT_OFF + (l * l + m) * 16 + bch] = acc[v];
      }
    }
  }

  // ---- step B: M(58x16) = CGall(58x25) @ CoutT(25x16), constant A tiles ----
#pragma unroll
  for (int mtile = 0; mtile < 4; ++mtile) {
    v8f acc = {0.f, 0.f, 0.f, 0.f, 0.f, 0.f, 0.f, 0.f};
#pragma unroll
    for (int kk = 0; kk < 7; ++kk) {
      const int ck = kk * 4 + 2 * h;
      v2f a; a.x = s_cg[(mtile * 16 + m) * CG_COLS + ck];
             a.y = s_cg[(mtile * 16 + m) * CG_COLS + ck + 1];
      v2f b; b.x = sw[COUT_OFF + ck * 16 + m];
             b.y = sw[COUT_OFF + (ck + 1) * 16 + m];
      acc = WMMA_F32X4(a, b, acc);
    }
#pragma unroll
    for (int v = 0; v < 8; ++v) {
      const int r = mtile * 16 + v + 8 * h;
      if (r < 58) sw[M_OFF + r * 16 + m] = acc[v];
    }
  }

  // ---- collapse + assemble 14x14 (replicates reference reshape order) ----
  {
    const int coffc[3] = {0, 48, 80};
    const int adim[3]  = {3, 2, 1};
    const int base3[9] = {0, 0, 0, 1, 4, 0, 13, 18, 33};
    for (int pos = lane; pos < 196; pos += 32) {
      const int row = pos / 14, col = pos - (pos / 14) * 14;
      int lr, rq, lc, cq;
      decode14(row, lr, rq);
      decode14(col, lc, cq);
      int L1 = lr, L2 = lc, RQ = rq, CQ = cq;
      if (lr < lc) { L1 = lc; L2 = lr; RQ = cq; CQ = rq; }  // transposed block
      const int A_ = adim[L1], B_ = adim[L2], N_ = 2 * L2 + 1;
      // reference reshape: flat = ((m*N + n)*A + a)*B + b over rows of N*B
      const int flat = RQ * (N_ * B_) + CQ;
      const int b_  = flat % B_;
      const int r1  = flat / B_;
      const int a_  = r1 % A_;
      const int r2  = r1 / A_;
      const int n_  = r2 % N_;
      const int m_  = r2 / N_;
      const int r58 = base3[L1 * 3 + L2] + m_ * N_ + n_;
      const float* cra = s_coll + coffc[L1] + a_ * 16;
      const float* crb = s_coll + coffc[L2] + b_ * 16;
      float s = 0.0f;
#pragma unroll
      for (int f = 0; f < 16; ++f)
        s += sw[M_OFF + r58 * 16 + f] * cra[f] * crb[f];
      out[(size_t)pair * 196 + pos] = s;
    }
  }
}

extern "C" void kernel_launch(void* const* d_in, const int* in_sizes, int n_in,
                              void* d_out, int out_size, void* d_ws, size_t ws_size,
                              hipStream_t stream) {
  const float* c0   = (const float*)d_in[0];
  const float* c1   = (const float*)d_in[1];
  const float* c2   = (const float*)d_in[2];
  const float* c3   = (const float*)d_in[3];
  const float* c4   = (const float*)d_in[4];
  const float* R    = (const float*)d_in[5];
  const float* wrad = (const float*)d_in[6];
  const float* wgt  = (const float*)d_in[7];
  const float* scol = (const float*)d_in[8];
  const float* pcol = (const float*)d_in[9];
  const float* dcol = (const float*)d_in[10];
  const int*   Z    = (const int*)d_in[11];
  const int*   prs  = (const int*)d_in[12];
  const int*   aix  = (const int*)d_in[13];
  const int    P    = in_sizes[13];

  float* cg = (float*)d_ws;  // 64*28 floats

  cg_setup_kernel<<<1, 64, 0, stream>>>(cg);

  const int blocks = (P + WAVES_PER_BLOCK - 1) / WAVES_PER_BLOCK;
  output_layer_kernel<<<blocks, BLOCK_THREADS, 0, stream>>>(
      c0, c1, c2, c3, c4, R, wrad, wgt, scol, pcol, dcol, Z, prs, aix,
      cg, (float*)d_out, P);
}